// GNNTransformer_300647711094
// MI455X (gfx1250) — compile-verified
//
#include <hip/hip_runtime.h>
#include <hip/hip_bf16.h>

#define HEADS 4
#define HDIM 32
#define FDIM 128            // HEADS*HDIM == Fin for both layers
#define NEG_SLOPE 0.2f
#define ENC_NEG_INF 0x007FFFFFu   // fenc(-inf)

typedef float v2f __attribute__((ext_vector_type(2)));
typedef float v8f __attribute__((ext_vector_type(8)));

// order-preserving float <-> uint mapping for atomicMax on floats
static __device__ __forceinline__ unsigned fenc(float f) {
    unsigned u = __float_as_uint(f);
    return (u & 0x80000000u) ? ~u : (u | 0x80000000u);
}
static __device__ __forceinline__ float fdec(unsigned u) {
    return (u & 0x80000000u) ? __uint_as_float(u & 0x7fffffffu)
                             : __uint_as_float(~u);
}

// ---------------------------------------------------------------------------
// GEMM: H[N x 128] = X[N x 128] * W[128 x 128], fp32 WMMA 16x16x4.
// One 16-row strip per block; 4 waves x 2 column tiles = 128 output cols.
// ---------------------------------------------------------------------------
#define XS_PITCH 132        // 128 + 4 pad -> conflict-free strided A reads
#define KCHUNK   32
#define WT_PITCH 36         // 32 + 4 pad  -> conflict-free strided B reads

__global__ __launch_bounds__(128, 4)
void gat_gemm_wmma(const float* __restrict__ X, const float* __restrict__ W,
                   float* __restrict__ Hout, int N)
{
    __shared__ float xs[16 * XS_PITCH];       // 16 rows x 128 K
    __shared__ float wt[FDIM * WT_PITCH];     // transposed chunk: [col][k]

    const int tid  = threadIdx.x;
    const int wave = tid >> 5;
    const int lane = tid & 31;
    const int j    = lane & 15;
    const int half = lane >> 4;
    const int row0 = blockIdx.x << 4;

    // stage X strip
    for (int i = tid; i < 16 * FDIM; i += 128) {
        int r = i >> 7, k = i & 127;
        int gr = row0 + r;
        xs[r * XS_PITCH + k] = (gr < N) ? X[(size_t)gr * FDIM + k] : 0.0f;
    }

    const int c0a = (wave * 2 + 0) * 16;
    const int c0b = (wave * 2 + 1) * 16;
    v8f acc0 = {0.f,0.f,0.f,0.f,0.f,0.f,0.f,0.f};
    v8f acc1 = {0.f,0.f,0.f,0.f,0.f,0.f,0.f,0.f};

    for (int kc = 0; kc < FDIM; kc += KCHUNK) {
        __syncthreads();
        // stage transposed W chunk [kc..kc+32) x 128  (coalesced global reads)
        for (int i = tid; i < KCHUNK * FDIM; i += 128) {
            int r = i >> 7, c = i & 127;
            wt[c * WT_PITCH + r] = W[(size_t)(kc + r) * FDIM + c];
        }
        __syncthreads();

        #pragma unroll
        for (int kk = 0; kk < KCHUNK; kk += 4) {
            const int k2 = kk + 2 * half;   // ISA A/B layout: lanes 16-31 hold K+2
            v2f a  = *(const v2f*)(xs + j * XS_PITCH + kc + k2);
            v2f b0 = *(const v2f*)(wt + (c0a + j) * WT_PITCH + k2);
            v2f b1 = *(const v2f*)(wt + (c0b + j) * WT_PITCH + k2);
            acc0 = __builtin_amdgcn_wmma_f32_16x16x4_f32(
                       false, a, false, b0, (short)0, acc0, false, false);
            acc1 = __builtin_amdgcn_wmma_f32_16x16x4_f32(
                       false, a, false, b1, (short)0, acc1, false, false);
        }
    }

    // C/D layout: VGPR v -> row v + 8*half, col = c0 + (lane&15)
    #pragma unroll
    for (int v = 0; v < 8; ++v) {
        int gr = row0 + v + 8 * half;
        if (gr < N) {
            Hout[(size_t)gr * FDIM + c0a + j] = acc0[v];
            Hout[(size_t)gr * FDIM + c0b + j] = acc1[v];
        }
    }
}

// ---------------------------------------------------------------------------
// alpha_s / alpha_d: per (node, head) dot over 32 channels. One wave == one
// head (HEAD_DIM == warpSize == 32 on CDNA5).
// ---------------------------------------------------------------------------
__global__ __launch_bounds__(128)
void gat_alpha(const float* __restrict__ Hm, const float* __restrict__ asrc,
               const float* __restrict__ adst, float* __restrict__ As,
               float* __restrict__ Ad, int N)
{
    int n  = blockIdx.x;
    int hd = threadIdx.x >> 5;
    int c  = threadIdx.x & 31;
    float hv = Hm[(size_t)n * FDIM + hd * HDIM + c];
    float s = hv * asrc[hd * HDIM + c];
    float d = hv * adst[hd * HDIM + c];
    #pragma unroll
    for (int off = 16; off > 0; off >>= 1) {
        s += __shfl_down(s, off, 32);
        d += __shfl_down(d, off, 32);
    }
    if (c == 0) {
        As[(size_t)n * HEADS + hd] = s;
        Ad[(size_t)n * HEADS + hd] = d;
    }
}

// zero the accumulator, init segment-max to -inf (encoded), denom to 0
__global__ void gat_init(float* __restrict__ agg, unsigned* __restrict__ emax,
                         float* __restrict__ denom, int N)
{
    long long t = (long long)blockIdx.x * blockDim.x + threadIdx.x;
    if (t < (long long)N * FDIM) agg[t] = 0.0f;
    if (t < (long long)N * HEADS) { emax[t] = ENC_NEG_INF; denom[t] = 0.0f; }
}

static __device__ __forceinline__ void edge_nodes(long long e, int E,
                                                  const int* __restrict__ src,
                                                  const int* __restrict__ dst,
                                                  int& s, int& d)
{
    if (e < (long long)E) { s = src[e]; d = dst[e]; }
    else                  { s = d = (int)(e - E); }   // self-loop
}

// pass 1: segment max of leaky_relu(alpha_s[src]+alpha_d[dst]) over dst
__global__ void gat_edge_max(const int* __restrict__ src, const int* __restrict__ dst,
                             const float* __restrict__ As, const float* __restrict__ Ad,
                             unsigned* __restrict__ emax, int E, int N)
{
    long long t = (long long)blockIdx.x * blockDim.x + threadIdx.x;
    if (t >= (long long)(E + N) * HEADS) return;
    int hd = (int)(t & (HEADS - 1));
    long long e = t >> 2;
    int s, d; edge_nodes(e, E, src, dst, s, d);
    float v = As[(size_t)s * HEADS + hd] + Ad[(size_t)d * HEADS + hd];
    v = (v > 0.0f) ? v : v * NEG_SLOPE;
    atomicMax(emax + (size_t)d * HEADS + hd, fenc(v));
}

// pass 2: segment sum of exp(e - emax[dst])
__global__ void gat_edge_denom(const int* __restrict__ src, const int* __restrict__ dst,
                               const float* __restrict__ As, const float* __restrict__ Ad,
                               const unsigned* __restrict__ emax,
                               float* __restrict__ denom, int E, int N)
{
    long long t = (long long)blockIdx.x * blockDim.x + threadIdx.x;
    if (t >= (long long)(E + N) * HEADS) return;
    int hd = (int)(t & (HEADS - 1));
    long long e = t >> 2;
    int s, d; edge_nodes(e, E, src, dst, s, d);
    float v = As[(size_t)s * HEADS + hd] + Ad[(size_t)d * HEADS + hd];
    v = (v > 0.0f) ? v : v * NEG_SLOPE;
    float ex = expf(v - fdec(emax[(size_t)d * HEADS + hd]));
    atomicAdd(denom + (size_t)d * HEADS + hd, ex);
}

// pass 3: out[dst] += h[src] * alpha   (one thread per (edge, feature))
__global__ void gat_edge_agg(const int* __restrict__ src, const int* __restrict__ dst,
                             const float* __restrict__ As, const float* __restrict__ Ad,
                             const unsigned* __restrict__ emax,
                             const float* __restrict__ denom,
                             const float* __restrict__ Hm, float* __restrict__ agg,
                             int E, int N)
{
    long long t = (long long)blockIdx.x * blockDim.x + threadIdx.x;
    if (t >= (long long)(E + N) * FDIM) return;
    int f = (int)(t & (FDIM - 1));
    long long e = t >> 7;
    int hd = f >> 5;
    int s, d; edge_nodes(e, E, src, dst, s, d);
    float v = As[(size_t)s * HEADS + hd] + Ad[(size_t)d * HEADS + hd];
    v = (v > 0.0f) ? v : v * NEG_SLOPE;
    float ex = expf(v - fdec(emax[(size_t)d * HEADS + hd]));
    float alpha = ex / denom[(size_t)d * HEADS + hd];
    atomicAdd(agg + (size_t)d * FDIM + f, Hm[(size_t)s * FDIM + f] * alpha);
}

// y = elu(y + bias)
__global__ void gat_bias_elu(float* __restrict__ y, const float* __restrict__ b,
                             long long total)
{
    long long t = (long long)blockIdx.x * blockDim.x + threadIdx.x;
    if (t >= total) return;
    float v = y[t] + b[t & (FDIM - 1)];
    y[t] = (v > 0.0f) ? v : expm1f(v);
}

// ---------------------------------------------------------------------------
extern "C" void kernel_launch(void* const* d_in, const int* in_sizes, int n_in,
                              void* d_out, int out_size, void* d_ws, size_t ws_size,
                              hipStream_t stream)
{
    const float* x   = (const float*)d_in[0];
    const int*   ei  = (const int*)  d_in[1];   // [2, E] int32
    const float* W1  = (const float*)d_in[2];
    const float* as1 = (const float*)d_in[3];
    const float* ad1 = (const float*)d_in[4];
    const float* b1  = (const float*)d_in[5];
    const float* W2  = (const float*)d_in[6];
    const float* as2 = (const float*)d_in[7];
    const float* ad2 = (const float*)d_in[8];
    const float* b2  = (const float*)d_in[9];
    float* out = (float*)d_out;

    const int N = in_sizes[0] / FDIM;
    const int E = in_sizes[1] / 2;
    const int* src = ei;
    const int* dst = ei + E;

    // workspace: h (N*128 f32) + As/Ad/emax/denom (N*4 each)  ~= 58 MB
    float*    h     = (float*)d_ws;
    float*    As    = h  + (size_t)N * FDIM;
    float*    Ad    = As + (size_t)N * HEADS;
    unsigned* emax  = (unsigned*)(Ad + (size_t)N * HEADS);
    float*    denom = (float*)(emax + (size_t)N * HEADS);

    const long long nf = (long long)N * FDIM;
    const long long eh = (long long)(E + N) * HEADS;
    const long long ef = (long long)(E + N) * FDIM;
    const int gemmGrid = (N + 15) / 16;
    const int nfGrid   = (int)((nf + 255) / 256);
    const int ehGrid   = (int)((eh + 255) / 256);
    const int efGrid   = (int)((ef + 255) / 256);

    for (int layer = 0; layer < 2; ++layer) {
        const float* Xin  = (layer == 0) ? x   : out;   // gemm consumes out
        const float* Wm   = (layer == 0) ? W1  : W2;    // before init re-zeros it
        const float* asrc = (layer == 0) ? as1 : as2;
        const float* adst = (layer == 0) ? ad1 : ad2;
        const float* bias = (layer == 0) ? b1  : b2;

        gat_gemm_wmma<<<gemmGrid, 128, 0, stream>>>(Xin, Wm, h, N);
        gat_alpha<<<N, 128, 0, stream>>>(h, asrc, adst, As, Ad, N);
        gat_init<<<nfGrid, 256, 0, stream>>>(out, emax, denom, N);
        gat_edge_max<<<ehGrid, 256, 0, stream>>>(src, dst, As, Ad, emax, E, N);
        gat_edge_denom<<<ehGrid, 256, 0, stream>>>(src, dst, As, Ad, emax, denom, E, N);
        gat_edge_agg<<<efGrid, 256, 0, stream>>>(src, dst, As, Ad, emax, denom, h, out, E, N);
        gat_bias_elu<<<nfGrid, 256, 0, stream>>>(out, bias, nf);
    }
}